// GCN_68736656606003
// MI455X (gfx1250) — compile-verified
//
#include <hip/hip_runtime.h>

typedef __attribute__((ext_vector_type(2))) float v2f;
typedef __attribute__((ext_vector_type(8))) float v8f;

#define N_NODES 100000
#define N_EDGES 1600000
#define D_FEAT  128
#define UNITS   128
#define M_TILES (N_NODES / 16)   // 6250, exact
#define N_TILES (UNITS / 16)     // 8

// ---------------- degree / normalization ----------------

__global__ void GCN_init_deg(float* __restrict__ deg) {
    int i = blockIdx.x * blockDim.x + threadIdx.x;
    if (i < N_NODES) deg[i] = 1.0f;          // self-loop weight (renorm trick)
}

__global__ void GCN_edge_deg(const long long* __restrict__ row,
                             const float* __restrict__ w,
                             float* __restrict__ deg) {
    int e = blockIdx.x * blockDim.x + threadIdx.x;
    if (e < N_EDGES) {
        int r = (int)row[e];
        unsafeAtomicAdd(&deg[r], w[e]);      // hardware global_atomic_add_f32
    }
}

__global__ void GCN_dinv(const float* __restrict__ deg,
                         float* __restrict__ dinv) {
    int i = blockIdx.x * blockDim.x + threadIdx.x;
    if (i < N_NODES) {
        float d = deg[i];
        dinv[i] = (d > 0.0f) ? rsqrtf(d) : 0.0f;
    }
}

// ---------------- dense transform: h = x @ kernel (fp32 WMMA) ----------------
// One wave per 16x16 output tile; K=128 in 32 steps of V_WMMA_F32_16X16X4_F32.
// A 16x4 f32: lanes 0-15 -> M=lane, K={0,1}; lanes 16-31 -> M=lane-16, K={2,3}
// (contiguous float2 per lane). B 4x16: lanes carry N, VGPR0/1 carry K pairs.
__global__ void __launch_bounds__(256)
GCN_gemm_wmma(const float* __restrict__ x,
              const float* __restrict__ kern,
              float* __restrict__ h) {
    const int lane = threadIdx.x & 31;
    const int wave = threadIdx.x >> 5;       // 0..7 -> n-tile
    const int m_tile = blockIdx.x;           // 0..6249
    if (m_tile >= M_TILES) return;           // wave-uniform (never taken)

    const int m0   = m_tile * 16;
    const int n0   = wave * 16;
    const int half = lane >> 4;              // 0: K pair {0,1}, 1: K pair {2,3}
    const int l16  = lane & 15;

    const float* arow = x    + (long long)(m0 + l16) * D_FEAT + half * 2;
    const float* bcol = kern + (half * 2) * UNITS + n0 + l16;

    v8f c = {};
    #pragma unroll 8
    for (int k = 0; k < D_FEAT; k += 4) {
        v2f a = *(const v2f*)(arow + k);             // x[m, k+2h], x[m, k+2h+1]
        v2f b;
        b.x = bcol[k * UNITS];                       // kernel[k+2h,   n]
        b.y = bcol[k * UNITS + UNITS];               // kernel[k+2h+1, n]
        c = __builtin_amdgcn_wmma_f32_16x16x4_f32(
                /*neg_a=*/false, a, /*neg_b=*/false, b,
                /*c_mod=*/(short)0, c, /*reuse_a=*/false, /*reuse_b=*/false);
    }

    // C/D layout: VGPR r -> M = m0 + r + 8*half, N = n0 + l16
    float* hp = h + (long long)(m0 + half * 8) * UNITS + n0 + l16;
    #pragma unroll
    for (int r = 0; r < 8; ++r) hp[r * UNITS] = c[r];
}

// ---------------- self-loop term + bias: out = bias + dinv^2 * h ----------------

__global__ void GCN_self(const float* __restrict__ h,
                         const float* __restrict__ dinv,
                         const float* __restrict__ bias,
                         float* __restrict__ out) {
    int idx = blockIdx.x * blockDim.x + threadIdx.x;
    if (idx < N_NODES * UNITS) {
        int i = idx >> 7;
        int f = idx & (UNITS - 1);
        float di = dinv[i];
        out[idx] = bias[f] + di * di * h[idx];
    }
}

// ---------------- edge aggregation: out[row] += nw * h[col] ----------------
// One wave per edge; each lane handles 4 features via f32 atomics (L2-resident).
__global__ void __launch_bounds__(256)
GCN_edge_agg(const long long* __restrict__ row,
             const long long* __restrict__ col,
             const float* __restrict__ w,
             const float* __restrict__ dinv,
             const float* __restrict__ h,
             float* __restrict__ out) {
    const int lane = threadIdx.x & 31;
    const int e = (blockIdx.x * 256 + threadIdx.x) >> 5;
    if (e >= N_EDGES) return;

    const int r = (int)row[e];
    const int c = (int)col[e];
    const float nw = w[e] * dinv[r] * dinv[c];

    const float* hc = h + (long long)c * UNITS;
    float* orow = out + (long long)r * UNITS;

    #pragma unroll
    for (int j = 0; j < 4; ++j) {
        int f = lane + j * 32;
        unsafeAtomicAdd(&orow[f], nw * hc[f]);
    }
}

// ---------------- launcher ----------------

extern "C" void kernel_launch(void* const* d_in, const int* in_sizes, int n_in,
                              void* d_out, int out_size, void* d_ws, size_t ws_size,
                              hipStream_t stream) {
    const float*     x    = (const float*)d_in[0];              // [N,128]
    const long long* ei   = (const long long*)d_in[1];          // [2,E] int64
    const float*     ew   = (const float*)d_in[2];              // [E]
    const float*     kern = (const float*)d_in[3];              // [128,128]
    const float*     bias = (const float*)d_in[4];              // [128]
    float* out = (float*)d_out;                                 // [N,128]

    const long long* row = ei;              // edge_index[0]
    const long long* col = ei + N_EDGES;    // edge_index[1]

    // workspace: h [N*128] | deg [N] | dinv [N]  (~52 MB)
    float* h    = (float*)d_ws;
    float* deg  = h + (size_t)N_NODES * UNITS;
    float* dinv = deg + N_NODES;

    GCN_init_deg<<<(N_NODES + 255) / 256, 256, 0, stream>>>(deg);
    GCN_edge_deg<<<(N_EDGES + 255) / 256, 256, 0, stream>>>(row, ew, deg);
    GCN_dinv<<<(N_NODES + 255) / 256, 256, 0, stream>>>(deg, dinv);

    GCN_gemm_wmma<<<M_TILES, 256, 0, stream>>>(x, kern, h);   // 8 waves = 8 n-tiles

    GCN_self<<<(N_NODES * UNITS + 255) / 256, 256, 0, stream>>>(h, dinv, bias, out);

    GCN_edge_agg<<<N_EDGES / 8, 256, 0, stream>>>(row, col, ew, dinv, h, out);
}